// GCN_53360673686003
// MI455X (gfx1250) — compile-verified
//
#include <hip/hip_runtime.h>

typedef float v2f __attribute__((ext_vector_type(2)));
typedef float v8f __attribute__((ext_vector_type(8)));

// ---------------------------------------------------------------------------
// Degree / norm kernels
// ---------------------------------------------------------------------------
__global__ void gcn_init_deg(float* __restrict__ deg, int n) {
    int i = blockIdx.x * blockDim.x + threadIdx.x;
    if (i < n) deg[i] = 1.0f;               // self-loop contributes 1
}

__global__ void gcn_accum_deg(const int* __restrict__ dst, float* __restrict__ deg, int e) {
    int i = blockIdx.x * blockDim.x + threadIdx.x;
    if (i < e) unsafeAtomicAdd(&deg[dst[i]], 1.0f);
}

__global__ void gcn_dinv(float* __restrict__ deg, int n) {
    int i = blockIdx.x * blockDim.x + threadIdx.x;
    if (i < n) deg[i] = rsqrtf(deg[i]);     // deg >= 1 always (self-loops)
}

__global__ void gcn_zero(float* __restrict__ p, int n) {
    int i = blockIdx.x * blockDim.x + threadIdx.x;
    if (i < n) p[i] = 0.0f;
}

// ---------------------------------------------------------------------------
// WMMA f32 GEMM: C[M,Ncols] = A[M,K] @ B[K,Ncols]
// One wave per 16x16 output tile; block = (Ncols/16) waves covering one M-tile.
// Uses V_WMMA_F32_16X16X4_F32, K-loop step 4.
// ---------------------------------------------------------------------------
__global__ void gcn_wmma_gemm_f32(const float* __restrict__ A,
                                  const float* __restrict__ B,
                                  float* __restrict__ C,
                                  int M, int K, int Ncols) {
    const int wave = threadIdx.x >> 5;
    const int lane = threadIdx.x & 31;
    const int row0 = blockIdx.x * 16;
    const int col0 = wave * 16;
    if (row0 >= M || col0 >= Ncols) return;

    const int m  = lane & 15;             // A row within tile / C column group
    const int kb = (lane >> 4) << 1;      // lane half selects K pair {0,1} or {2,3}
    const int n  = lane & 15;             // B/C column within tile

    const int mrow = (row0 + m < M) ? (row0 + m) : (M - 1);
    const float* Arow = A + (size_t)mrow * K;

    v8f c = {};
    for (int k0 = 0; k0 < K; k0 += 4) {
        // A 16x4 tile: v2f per lane, contiguous, 8B-aligned
        v2f a = *(const v2f*)(Arow + k0 + kb);
        // B 4x16 tile: lane half selects K pair, strided by Ncols
        v2f b;
        b.x = B[(size_t)(k0 + kb) * Ncols + col0 + n];
        b.y = B[(size_t)(k0 + kb + 1) * Ncols + col0 + n];
        c = __builtin_amdgcn_wmma_f32_16x16x4_f32(
                /*neg_a=*/false, a, /*neg_b=*/false, b,
                /*c_mod=*/(short)0, c, /*reuse_a=*/false, /*reuse_b=*/false);
    }

    // C/D layout: VGPR r -> M=r (lanes 0-15), M=r+8 (lanes 16-31)
    const int rowBase = row0 + ((lane >> 4) << 3);
    const int col = col0 + n;
#pragma unroll
    for (int r = 0; r < 8; ++r) {
        int row = rowBase + r;
        if (row < M) C[(size_t)row * Ncols + col] = c[r];
    }
}

// ---------------------------------------------------------------------------
// Edge scatter: one wave per edge. Lane l carries features [l*V, l*V+V).
// F=128 -> float4 per lane; F=64 -> float2 per lane.
// agg[dst] += hW[src] * (dinv[src]*dinv[dst])
// ---------------------------------------------------------------------------
__global__ void gcn_scatter_f128(const int* __restrict__ src, const int* __restrict__ dst,
                                 const float* __restrict__ dinv,
                                 const float* __restrict__ h, float* __restrict__ agg,
                                 int E) {
    int t = blockIdx.x * blockDim.x + threadIdx.x;
    int e = t >> 5;
    int lane = t & 31;
    if (e >= E) return;
    int s = src[e], d = dst[e];
    float w = dinv[s] * dinv[d];
    const float4* hv = (const float4*)(h + (size_t)s * 128);
    float4 v = hv[lane];
    float* out = agg + (size_t)d * 128 + lane * 4;
    unsafeAtomicAdd(out + 0, v.x * w);
    unsafeAtomicAdd(out + 1, v.y * w);
    unsafeAtomicAdd(out + 2, v.z * w);
    unsafeAtomicAdd(out + 3, v.w * w);
}

__global__ void gcn_scatter_f64(const int* __restrict__ src, const int* __restrict__ dst,
                                const float* __restrict__ dinv,
                                const float* __restrict__ h, float* __restrict__ agg,
                                int E) {
    int t = blockIdx.x * blockDim.x + threadIdx.x;
    int e = t >> 5;
    int lane = t & 31;
    if (e >= E) return;
    int s = src[e], d = dst[e];
    float w = dinv[s] * dinv[d];
    const float2* hv = (const float2*)(h + (size_t)s * 64);
    float2 v = hv[lane];
    float* out = agg + (size_t)d * 64 + lane * 2;
    unsafeAtomicAdd(out + 0, v.x * w);
    unsafeAtomicAdd(out + 1, v.y * w);
}

// ---------------------------------------------------------------------------
// Finalize: out = act(agg + hW*dinv^2 (self-loop) + bias); F = 1<<logF
// ---------------------------------------------------------------------------
__global__ void gcn_finalize(const float* __restrict__ agg, const float* __restrict__ h,
                             const float* __restrict__ dinv, const float* __restrict__ bias,
                             float* __restrict__ out, int total, int logF, int do_relu) {
    int t = blockIdx.x * blockDim.x + threadIdx.x;
    if (t >= total) return;
    int i = t >> logF;
    int f = t & ((1 << logF) - 1);
    float di = dinv[i];
    float v = agg[t] + h[t] * di * di + bias[f];
    if (do_relu) v = fmaxf(v, 0.0f);
    out[t] = v;
}

// ---------------------------------------------------------------------------
extern "C" void kernel_launch(void* const* d_in, const int* in_sizes, int n_in,
                              void* d_out, int out_size, void* d_ws, size_t ws_size,
                              hipStream_t stream) {
    const float* x   = (const float*)d_in[0];
    const int*   ei  = (const int*)d_in[1];     // [2,E] flat: src then dst
    const float* W1  = (const float*)d_in[2];
    const float* b1  = (const float*)d_in[3];
    const float* W2  = (const float*)d_in[4];
    const float* b2  = (const float*)d_in[5];
    float* out = (float*)d_out;

    const int D_IN  = 128;
    const int D_H   = 128;
    const int D_OUT = 64;
    const int N = in_sizes[0] / D_IN;
    const int E = in_sizes[1] / 2;
    const int* e_src = ei;
    const int* e_dst = ei + E;

    // Workspace layout (floats): dinv[N] | hW[N*128] | agg[N*128]
    float* ws   = (float*)d_ws;
    float* dinv = ws;
    float* hW   = ws + (((size_t)N + 3) & ~(size_t)3);
    float* agg  = hW + (size_t)N * D_H;

    const int TB = 256;
    auto cdiv = [](long long a, long long b) { return (int)((a + b - 1) / b); };

    // 1) symmetric normalization coefficients
    gcn_init_deg<<<cdiv(N, TB), TB, 0, stream>>>(dinv, N);
    gcn_accum_deg<<<cdiv(E, TB), TB, 0, stream>>>(e_dst, dinv, E);
    gcn_dinv<<<cdiv(N, TB), TB, 0, stream>>>(dinv, N);

    // 2) layer 1: hW = x @ W1  (WMMA), then scatter + finalize(ReLU)
    gcn_zero<<<cdiv((long long)N * D_H, TB), TB, 0, stream>>>(agg, N * D_H);
    gcn_wmma_gemm_f32<<<cdiv(N, 16), (D_H / 16) * 32, 0, stream>>>(x, W1, hW, N, D_IN, D_H);
    gcn_scatter_f128<<<cdiv((long long)E * 32, TB), TB, 0, stream>>>(e_src, e_dst, dinv, hW, agg, E);
    gcn_finalize<<<cdiv((long long)N * D_H, TB), TB, 0, stream>>>(agg, hW, dinv, b1,
                                                                  agg /*x1 in place*/,
                                                                  N * D_H, 7, 1);

    // 3) layer 2: hW2 = x1 @ W2 (WMMA, reuse hW buffer), scatter into d_out, finalize
    gcn_wmma_gemm_f32<<<cdiv(N, 16), (D_OUT / 16) * 32, 0, stream>>>(agg, W2, hW, N, D_H, D_OUT);
    gcn_zero<<<cdiv((long long)N * D_OUT, TB), TB, 0, stream>>>(out, N * D_OUT);
    gcn_scatter_f64<<<cdiv((long long)E * 32, TB), TB, 0, stream>>>(e_src, e_dst, dinv, hW, out, E);
    gcn_finalize<<<cdiv((long long)N * D_OUT, TB), TB, 0, stream>>>(out, hW, dinv, b2, out,
                                                                    N * D_OUT, 6, 0);
}